// CrossNet_13322988552563
// MI455X (gfx1250) — compile-verified
//
#include <hip/hip_runtime.h>

// CrossNet (DCN) fused 3-layer kernel for MI455X (gfx1250, wave32).
// Per row r, per layer: s = dot(y_r, w);  y_r = y_r*(s+1) + b.
// Memory-bound: 128MB in + 128MB out -> ~11us at 23.3 TB/s. All three layers
// fused; row lives in VGPRs; weights/biases staged to LDS with the CDNA5
// async global->LDS path (ASYNCcnt), rows streamed with non-temporal hints.

#define DD 2048                      // feature dim
#define V4_PER_ROW (DD / 4)          // 512 float4 per row
#define V4_PER_LANE (V4_PER_ROW / 32)// 16 float4 per lane (64 floats)
#define WAVES_PER_BLOCK 8
#define THREADS 256

typedef float __attribute__((ext_vector_type(4))) f4;

// Payload type the async-LDS builtin expects (per hipcc diagnostic):
// 'int __attribute__((vector_size(16)))'
typedef int v4i __attribute__((vector_size(16)));
typedef __attribute__((address_space(1))) v4i* gv4i_p;  // global
typedef __attribute__((address_space(3))) v4i* lv4i_p;  // LDS

#if __has_builtin(__builtin_amdgcn_global_load_async_to_lds_b128)
#define HAVE_ASYNC_LDS 1
#else
#define HAVE_ASYNC_LDS 0
#endif

__global__ __launch_bounds__(THREADS) void crossnet3_fused(
    const float* __restrict__ x,
    const float* __restrict__ w1, const float* __restrict__ b1,
    const float* __restrict__ w2, const float* __restrict__ b2,
    const float* __restrict__ w3, const float* __restrict__ b3,
    float* __restrict__ out)
{
    __shared__ float s_wb[6 * DD];   // w1,b1,w2,b2,w3,b3 : 48 KB

    const int tid  = threadIdx.x;
    const int lane = tid & 31;
    const int wave = tid >> 5;
    const size_t row = (size_t)blockIdx.x * WAVES_PER_BLOCK + (size_t)wave;

    // ---- Stage weights/biases into LDS (async copy path on CDNA5) ----
    // 6 arrays x 512 float4; 256 threads -> exactly 2 chunks per thread per
    // array. Fully straight-line (no data-dependent bounds -> no exec-mask
    // branches in the prologue).
    const float* srcs[6] = { w1, b1, w2, b2, w3, b3 };
    #pragma unroll
    for (int a = 0; a < 6; ++a) {
        const f4* gs = (const f4*)srcs[a];
        f4*       ls = (f4*)(s_wb + a * DD);
#if HAVE_ASYNC_LDS
        __builtin_amdgcn_global_load_async_to_lds_b128(
            (gv4i_p)(gs + tid),          (lv4i_p)(ls + tid),          0, 0);
        __builtin_amdgcn_global_load_async_to_lds_b128(
            (gv4i_p)(gs + tid + THREADS),(lv4i_p)(ls + tid + THREADS),0, 0);
#else
        ls[tid]           = gs[tid];
        ls[tid + THREADS] = gs[tid + THREADS];
#endif
    }

    // Overlap: prefetch this wave's row cachelines while weights stage.
    const f4* rowp = (const f4*)(x + row * DD);
    #pragma unroll
    for (int j = 0; j < V4_PER_LANE; ++j)
        __builtin_prefetch((const void*)(rowp + j * 32 + lane), 0, 0);

#if HAVE_ASYNC_LDS
    asm volatile("s_wait_asynccnt 0" ::: "memory");
#endif
    __syncthreads();

    // ---- Load row into registers (read-once -> non-temporal) ----
    f4 y[V4_PER_LANE];
    #pragma unroll
    for (int j = 0; j < V4_PER_LANE; ++j)
        y[j] = __builtin_nontemporal_load(rowp + j * 32 + lane);

    // ---- Three fused cross layers ----
    #pragma unroll
    for (int layer = 0; layer < 3; ++layer) {
        const f4* lw = (const f4*)(s_wb + (2 * layer + 0) * DD);
        const f4* lb = (const f4*)(s_wb + (2 * layer + 1) * DD);

        float acc = 0.0f;
        #pragma unroll
        for (int j = 0; j < V4_PER_LANE; ++j) {
            const f4 wv = lw[j * 32 + lane];
            acc = fmaf(y[j].x, wv.x, acc);
            acc = fmaf(y[j].y, wv.y, acc);
            acc = fmaf(y[j].z, wv.z, acc);
            acc = fmaf(y[j].w, wv.w, acc);
        }
        // wave32 all-reduce (EXEC all ones; no divergence in this kernel)
        #pragma unroll
        for (int off = 16; off >= 1; off >>= 1)
            acc += __shfl_xor(acc, off, 32);

        const float t = acc + 1.0f;  // y*(s+1) + b  ==  y*s + b + y
        #pragma unroll
        for (int j = 0; j < V4_PER_LANE; ++j) {
            const f4 bv = lb[j * 32 + lane];
            y[j].x = fmaf(y[j].x, t, bv.x);
            y[j].y = fmaf(y[j].y, t, bv.y);
            y[j].z = fmaf(y[j].z, t, bv.z);
            y[j].w = fmaf(y[j].w, t, bv.w);
        }
    }

    // ---- Store result (write-once -> non-temporal) ----
    f4* outp = (f4*)(out + row * DD);
    #pragma unroll
    for (int j = 0; j < V4_PER_LANE; ++j)
        __builtin_nontemporal_store(y[j], outp + j * 32 + lane);
}

extern "C" void kernel_launch(void* const* d_in, const int* in_sizes, int n_in,
                              void* d_out, int out_size, void* d_ws, size_t ws_size,
                              hipStream_t stream) {
    const float* x  = (const float*)d_in[0];
    const float* w1 = (const float*)d_in[1];
    const float* b1 = (const float*)d_in[2];
    const float* w2 = (const float*)d_in[3];
    const float* b2 = (const float*)d_in[4];
    const float* w3 = (const float*)d_in[5];
    const float* b3 = (const float*)d_in[6];
    float* out = (float*)d_out;

    const int B = in_sizes[0] / DD;              // 16384
    const int grid = B / WAVES_PER_BLOCK;        // 2048 blocks * 8 waves = 1 wave/row

    crossnet3_fused<<<grid, THREADS, 0, stream>>>(x, w1, b1, w2, b2, w3, b3, out);
}